// EFR_61272003445297
// MI455X (gfx1250) — compile-verified
//
#include <hip/hip_runtime.h>

typedef __attribute__((ext_vector_type(2))) float v2f;
typedef __attribute__((ext_vector_type(8))) float v8f;

// out[n,k,o] = sum_d feat[n,k,d]*(W[o,d] + W[o,64]/64)
//            + sum_d x1[n,d]  *(W[o,65+d] - W[o,d])
// => single GEMM: A_ext[1.6M x 128] @ B_ext[128 x 128]
//    A_ext row = [ x[idx[n,k]][0:64] | x[idx[n,0]][0:64] ]
//    B_ext in LDS, o-major [o][d2] stride 132: B-pair = one ds_load_b64,
//    full-wave bank-conflict-free (banks (4o+d)%64 cover all 64 exactly once).
#define BSTRIDE 132

__global__ __launch_bounds__(128) void efr_wmma_kernel(
    const float* __restrict__ x,    // [NS, 64]
    const int*   __restrict__ idx,  // [NQ, 32]
    const float* __restrict__ W,    // [128, 129]
    float*       __restrict__ out,  // [NQ, 32, 128]
    int ntiles)                     // (NQ*32)/16
{
  __shared__ float ldsW[128 * BSTRIDE];   // 67.5 KB (gfx1250: 320KB/WG)

  // ---- Build B_ext in LDS: one row o per thread, contiguous writes ----
  {
    int o = threadIdx.x;              // 0..127
    const float* wrow = W + o * 129;
    float*       lrow = ldsW + o * BSTRIDE;
    float w2s = wrow[64] * 0.015625f; // w2/64 folds the fmean channel
#pragma unroll 8
    for (int d = 0; d < 64; ++d)
      lrow[d] = wrow[d] + w2s;                  // Weff[o][d]
#pragma unroll 8
    for (int d = 0; d < 64; ++d)
      lrow[64 + d] = wrow[65 + d] - wrow[d];    // Wb[o][d]
  }
  __syncthreads();

  int lane = threadIdx.x & 31;
  int wid  = threadIdx.x >> 5;
  int tile = blockIdx.x * 4 + wid;
  if (tile >= ntiles) return;

  int rowBase = tile * 16;      // 16 output rows, all same query n
  int n = rowBase >> 5;         // K=32 rows per query, 16 | 32
  int m = lane & 15;            // A row within tile
  int r = rowBase + m;
  int nb = idx[n * 32 + (r & 31)];  // this row's neighbor
  int c0 = idx[n * 32];             // center point index (idx[n,0])
  const float* arow = x + (long)nb * 64;
  const float* xrow = x + (long)c0 * 64;
  int koff = (lane >> 4) << 1;  // ISA A-layout: lanes>=16 hold K%4 in {2,3}

  // ---- Load A_ext fragment: 32 K-chunks of 4, float2 per lane per chunk ----
  v2f a[32];
#pragma unroll
  for (int j = 0; j < 16; ++j)
    a[j] = *(const v2f*)(arow + 4 * j + koff);      // feature half
#pragma unroll
  for (int j = 0; j < 16; ++j)
    a[16 + j] = *(const v2f*)(xrow + 4 * j + koff); // x1 half (bias GEMM)

  int ocol  = lane & 15;
  int mhalf = (lane >> 4) << 3; // D-layout: lanes>=16 hold rows M=8..15

  // ---- 8 N-tiles x 32 K-steps of v_wmma_f32_16x16x4_f32 ----
#pragma unroll
  for (int t = 0; t < 8; ++t) {
    int o = t * 16 + ocol;
    const float* bptr = ldsW + o * BSTRIDE + koff; // 8B-aligned
    v8f acc = {0.f, 0.f, 0.f, 0.f, 0.f, 0.f, 0.f, 0.f};
#pragma unroll
    for (int c = 0; c < 32; ++c) {
      v2f b = *(const v2f*)(bptr + 4 * c);         // one ds_load_b64
      acc = __builtin_amdgcn_wmma_f32_16x16x4_f32(
          /*neg_a=*/false, a[c], /*neg_b=*/false, b,
          /*c_mod=*/(short)0, acc, /*reuse_a=*/false, /*reuse_b=*/false);
    }
    // store 16x16 D tile: lane's column o, 8 rows; non-temporal (write-once
    // stream, keep L2 for the x-gather working set)
    float* op = out + (long)(rowBase + mhalf) * 128 + o;
#pragma unroll
    for (int v = 0; v < 8; ++v)
      __builtin_nontemporal_store(acc[v], op + (long)v * 128);
  }
}

extern "C" void kernel_launch(void* const* d_in, const int* in_sizes, int n_in,
                              void* d_out, int out_size, void* d_ws, size_t ws_size,
                              hipStream_t stream) {
  const float* x   = (const float*)d_in[0];
  const int*   idx = (const int*)d_in[1];
  const float* W   = (const float*)d_in[2];
  float* out = (float*)d_out;

  int rows   = in_sizes[1];          // NQ * 32 gathered rows
  int ntiles = rows / 16;            // 16 rows per wave tile
  int nblk   = (ntiles + 3) / 4;     // 4 waves (128 threads) per block

  efr_wmma_kernel<<<nblk, 128, 0, stream>>>(x, idx, W, out, ntiles);
}